// HypergraphClassifier_3298534884164
// MI455X (gfx1250) — compile-verified
//
#include <hip/hip_runtime.h>
#include <hip/hip_bf16.h>

// ---------------------------------------------------------------------------
// HypergraphConv classifier for MI455X (gfx1250, wave32).
//   out = head( meanpool( relu(conv2( relu(conv1(x)) )) ) )
//   conv(x) = Dinv * H * (Binv * (H^T * (x @ W))) + b
// GEMMs: fp32 WMMA (v_wmma_f32_16x16x4_f32), W pair-interleaved in LDS so each
// B fragment is a single contiguous ds_load_b64. Scatters: native fp32 global
// atomics (agent scope, relaxed). Pooling: LDS ds_add_f32 reduction.
// ---------------------------------------------------------------------------

#define N_NODES 100000
#define N_EDGES 20000
#define N_NNZ   1600000
#define N_G     64
#define DF      128   // feature dim for both layers

typedef __attribute__((ext_vector_type(2))) float v2f;
typedef __attribute__((ext_vector_type(8))) float v8f;

// Native, non-returning fp32 atomic adds (relaxed, device scope -> VMEM fp add;
// workgroup scope on LDS pointers -> ds_add_f32).
__device__ __forceinline__ void atomAddDev(float* p, float v) {
  __hip_atomic_fetch_add(p, v, __ATOMIC_RELAXED, __HIP_MEMORY_SCOPE_AGENT);
}
__device__ __forceinline__ void atomAddWg(float* p, float v) {
  __hip_atomic_fetch_add(p, v, __ATOMIC_RELAXED, __HIP_MEMORY_SCOPE_WORKGROUP);
}

// --------------------------- zero (float4) ---------------------------------
__global__ void zero_f32x4(float* __restrict__ p, int n4) {
  int i = blockIdx.x * blockDim.x + threadIdx.x;
  if (i < n4) ((float4*)p)[i] = make_float4(0.f, 0.f, 0.f, 0.f);
}

// --------------------------- degrees ---------------------------------------
__global__ void degrees_kernel(const int* __restrict__ src, const int* __restrict__ dst,
                               float* __restrict__ deg_n, float* __restrict__ deg_e, int nnz) {
  int i = blockIdx.x * blockDim.x + threadIdx.x;
  if (i < nnz) {
    atomAddDev(&deg_n[src[i]], 1.0f);
    atomAddDev(&deg_e[dst[i]], 1.0f);
  }
}

__global__ void count_nodes_kernel(const int* __restrict__ batch, float* __restrict__ cnt, int n) {
  int i = blockIdx.x * blockDim.x + threadIdx.x;
  if (i < n) atomAddDev(&cnt[batch[i]], 1.0f);
}

__global__ void reciprocal_kernel(float* __restrict__ p, int n) {
  int i = blockIdx.x * blockDim.x + threadIdx.x;
  if (i < n) {
    float v = p[i];
    p[i] = (v > 0.f) ? (1.0f / v) : 0.f;
  }
}

// --------------------------- fp32 WMMA GEMM --------------------------------
// H[N,128] = X[N,128] @ W[128,128]   (row major, N multiple of 16)
// Block = 256 threads = 8 waves; wave w computes rows [16*rt, 16*rt+16) x all
// 128 columns (8 tiles of 16) with V_WMMA_F32_16X16X4_F32 over K=128 (32 steps).
// A fragment (16x4 f32): lanes 0-15 -> K=k,k+1 ; lanes 16-31 -> K=k+2,k+3.
// B fragment (4x16 f32): VGPR0 = rows K0/K2, VGPR1 = rows K1/K3 (mirrors A).
// W is staged pair-interleaved in LDS: Ws[(k>>1)*256 + n*2 + (k&1)] = W[k][n],
// so each lane's (b.x, b.y) = (W[ka][n], W[ka+1][n]) is one 8-byte ds_load_b64
// at immediate offsets t*128B -- no register repacking in the inner loop.
__global__ __launch_bounds__(256) void gemm128_wmma_f32(
    const float* __restrict__ X, const float* __restrict__ W,
    float* __restrict__ H, int nTiles) {
  __shared__ float Ws[DF * DF];          // 64 KB of the 320 KB WGP LDS
  for (int i = threadIdx.x; i < DF * DF; i += 256) {
    const int k = i >> 7, n = i & 127;
    Ws[((k >> 1) << 8) + (n << 1) + (k & 1)] = W[i];
  }
  __syncthreads();

  const int wave = threadIdx.x >> 5;
  const int lane = threadIdx.x & 31;
  const int rt   = blockIdx.x * 8 + wave;
  if (rt >= nTiles) return;              // wave-uniform; EXEC stays all-ones

  const int half = lane >> 4;            // 0: lanes 0-15, 1: lanes 16-31
  const int l    = lane & 15;
  const float* xrow = X + (rt * 16 + l) * DF;

  v8f acc[8] = {};
  for (int k = 0; k < DF; k += 4) {
    const int ka = k + 2 * half;                       // even
    v2f a = *(const v2f*)(xrow + ka);                  // A[M=l][K=ka..ka+1]
    const float* brow = Ws + ((ka >> 1) << 8) + (l << 1);
#pragma unroll
    for (int t = 0; t < 8; ++t) {
      v2f b = *(const v2f*)(brow + t * 32);            // (W[ka][n], W[ka+1][n])
      acc[t] = __builtin_amdgcn_wmma_f32_16x16x4_f32(
          /*neg_a=*/false, a, /*neg_b=*/false, b,
          /*c_mod=*/(short)0, acc[t], /*reuse_a=*/false, /*reuse_b=*/false);
    }
  }
  // C/D layout: VGPR r holds M = r + 8*half, N = l  (ISA 7.12.2)
#pragma unroll
  for (int t = 0; t < 8; ++t) {
#pragma unroll
    for (int r = 0; r < 8; ++r) {
      const int M = r + 8 * half;
      H[(rt * 16 + M) * DF + t * 16 + l] = acc[t][r];
    }
  }
}

// --------------------------- scatters --------------------------------------
// edge[d] += h[s]   (one wave per incidence, float4 per lane, 4 atomics).
// e is wave-uniform: force it scalar so src/dst come in via s_load.
__global__ __launch_bounds__(256) void scatter_node_to_edge(
    const float* __restrict__ h, const int* __restrict__ src,
    const int* __restrict__ dst, float* __restrict__ edge, int nnz) {
  int e = blockIdx.x * 8 + (threadIdx.x >> 5);
  if (e >= nnz) return;
  e = __builtin_amdgcn_readfirstlane(e);
  const int lane = threadIdx.x & 31;
  const int s = src[e], d = dst[e];
  const float4 v = *(const float4*)(h + (long)s * DF + lane * 4);
  float* p = edge + (long)d * DF + lane * 4;
  atomAddDev(p + 0, v.x); atomAddDev(p + 1, v.y);
  atomAddDev(p + 2, v.z); atomAddDev(p + 3, v.w);
}

// out[s] += Binv[d] * edge[d]
__global__ __launch_bounds__(256) void scatter_edge_to_node(
    const float* __restrict__ edge, const float* __restrict__ binv,
    const int* __restrict__ src, const int* __restrict__ dst,
    float* __restrict__ out, int nnz) {
  int e = blockIdx.x * 8 + (threadIdx.x >> 5);
  if (e >= nnz) return;
  e = __builtin_amdgcn_readfirstlane(e);
  const int lane = threadIdx.x & 31;
  const int s = src[e], d = dst[e];
  const float bi = binv[d];
  const float4 v = *(const float4*)(edge + (long)d * DF + lane * 4);
  float* p = out + (long)s * DF + lane * 4;
  atomAddDev(p + 0, bi * v.x); atomAddDev(p + 1, bi * v.y);
  atomAddDev(p + 2, bi * v.z); atomAddDev(p + 3, bi * v.w);
}

// --------------------------- finalize: relu(Dinv*acc + b) -------------------
__global__ __launch_bounds__(256) void finalize_relu(
    const float* __restrict__ acc, const float* __restrict__ dinv,
    const float* __restrict__ bias, float* __restrict__ out, int total) {
  const int idx = blockIdx.x * blockDim.x + threadIdx.x;
  const int base = idx * 4;
  if (base >= total) return;
  const int n = base >> 7;          // /128
  const int f = base & 127;
  const float di = dinv[n];
  float4 v = *(const float4*)(acc + base);
  const float4 b = *(const float4*)(bias + f);
  v.x = fmaxf(di * v.x + b.x, 0.f);
  v.y = fmaxf(di * v.y + b.y, 0.f);
  v.z = fmaxf(di * v.z + b.z, 0.f);
  v.w = fmaxf(di * v.w + b.w, 0.f);
  *(float4*)(out + base) = v;
}

// Layer-2 finalize fused with mean-pool accumulation.
// 16 nodes per block; `batch` is sorted, so blocks are almost always
// batch-uniform -> reduce in LDS (ds_add_f32), 128 global atomics per block.
__global__ __launch_bounds__(256) void finalize_relu_pool(
    const float* __restrict__ acc, const float* __restrict__ dinv,
    const float* __restrict__ bias, const int* __restrict__ batch,
    float* __restrict__ pooled, int nNodes) {
  __shared__ float accs[DF];
  const int nbase = blockIdx.x * 16;
  const int g0 = batch[nbase];
  const bool uniform = (g0 == batch[nbase + 15]);
  if (threadIdx.x < DF) accs[threadIdx.x] = 0.f;
  __syncthreads();
  const int f = threadIdx.x & 127;
  const float bb = bias[f];
#pragma unroll
  for (int p = 0; p < 8; ++p) {
    const int n = nbase + p * 2 + (threadIdx.x >> 7);
    float v = fmaxf(dinv[n] * acc[(long)n * DF + f] + bb, 0.f);
    if (uniform) atomAddWg(&accs[f], v);                   // ds_add_f32
    else         atomAddDev(&pooled[batch[n] * DF + f], v); // rare boundary
  }
  __syncthreads();
  if (uniform && threadIdx.x < DF)
    atomAddDev(&pooled[g0 * DF + threadIdx.x], accs[threadIdx.x]);
}

// --------------------------- head: (pooled/cnt) @ Wl + bl -------------------
__global__ __launch_bounds__(128) void head_kernel(
    const float* __restrict__ pooled, const float* __restrict__ cnt,
    const float* __restrict__ Wl, const float* __restrict__ bl,
    float* __restrict__ out) {
  const int t = threadIdx.x;            // 128 = G(64) * D_OUT(2)
  const int g = t >> 1, o = t & 1;
  float s = 0.f;
#pragma unroll 8
  for (int f = 0; f < DF; ++f) s += pooled[g * DF + f] * Wl[f * 2 + o];
  out[t] = s / fmaxf(cnt[g], 1.0f) + bl[o];
}

// ---------------------------------------------------------------------------
extern "C" void kernel_launch(void* const* d_in, const int* in_sizes, int n_in,
                              void* d_out, int out_size, void* d_ws, size_t ws_size,
                              hipStream_t stream) {
  (void)in_sizes; (void)n_in; (void)out_size; (void)ws_size;
  const float* x   = (const float*)d_in[0];
  const int* hei   = (const int*)d_in[1];
  const int* src   = hei;               // node ids [NNZ]
  const int* dst   = hei + N_NNZ;       // hyperedge ids [NNZ]
  const int* batch = (const int*)d_in[2];
  const float* W1  = (const float*)d_in[3];
  const float* b1  = (const float*)d_in[4];
  const float* W2  = (const float*)d_in[5];
  const float* b2  = (const float*)d_in[6];
  const float* Wl  = (const float*)d_in[7];
  const float* bl  = (const float*)d_in[8];

  // workspace carve-out (floats); total ~113 MB
  float* ws     = (float*)d_ws;
  float* bufH   = ws;                         // N*DF
  float* bufO   = bufH + (long)N_NODES * DF;  // N*DF
  float* edge   = bufO + (long)N_NODES * DF;  // E*DF
  float* dinv   = edge + (long)N_EDGES * DF;  // N
  float* binv   = dinv + N_NODES;             // E
  float* pooled = binv + N_EDGES;             // G*DF
  float* cnt    = pooled + N_G * DF;          // G

  const int T = 256;
  const int nTiles   = N_NODES / 16;                 // 6250
  const int gemmBlk  = (nTiles + 7) / 8;             // 782
  const int scatBlk  = (N_NNZ + 7) / 8;              // 200000
  const int finBlk   = (N_NODES * DF / 4 + T - 1) / T;

  // degrees + counts + pooled accumulators (contiguous zero region)
  const int statN = N_NODES + N_EDGES + N_G * DF + N_G;   // multiple of 4
  zero_f32x4<<<(statN / 4 + T - 1) / T, T, 0, stream>>>(dinv, statN / 4);
  degrees_kernel<<<(N_NNZ + T - 1) / T, T, 0, stream>>>(src, dst, dinv, binv, N_NNZ);
  count_nodes_kernel<<<(N_NODES + T - 1) / T, T, 0, stream>>>(batch, cnt, N_NODES);
  reciprocal_kernel<<<(N_NODES + T - 1) / T, T, 0, stream>>>(dinv, N_NODES);
  reciprocal_kernel<<<(N_EDGES + T - 1) / T, T, 0, stream>>>(binv, N_EDGES);

  // ---- layer 1: h = relu(Dinv H Binv H^T (x@W1) + b1) -> bufH
  gemm128_wmma_f32<<<gemmBlk, T, 0, stream>>>(x, W1, bufH, nTiles);
  zero_f32x4<<<(N_EDGES * DF / 4 + T - 1) / T, T, 0, stream>>>(edge, N_EDGES * DF / 4);
  scatter_node_to_edge<<<scatBlk, T, 0, stream>>>(bufH, src, dst, edge, N_NNZ);
  zero_f32x4<<<(N_NODES * DF / 4 + T - 1) / T, T, 0, stream>>>(bufO, N_NODES * DF / 4);
  scatter_edge_to_node<<<scatBlk, T, 0, stream>>>(edge, binv, src, dst, bufO, N_NNZ);
  finalize_relu<<<finBlk, T, 0, stream>>>(bufO, dinv, b1, bufH, N_NODES * DF);

  // ---- layer 2 (GEMM out -> bufO, node accum -> bufH), fused pooling
  gemm128_wmma_f32<<<gemmBlk, T, 0, stream>>>(bufH, W2, bufO, nTiles);
  zero_f32x4<<<(N_EDGES * DF / 4 + T - 1) / T, T, 0, stream>>>(edge, N_EDGES * DF / 4);
  scatter_node_to_edge<<<scatBlk, T, 0, stream>>>(bufO, src, dst, edge, N_NNZ);
  zero_f32x4<<<(N_NODES * DF / 4 + T - 1) / T, T, 0, stream>>>(bufH, N_NODES * DF / 4);
  scatter_edge_to_node<<<scatBlk, T, 0, stream>>>(edge, binv, src, dst, bufH, N_NNZ);
  finalize_relu_pool<<<N_NODES / 16, T, 0, stream>>>(bufH, dinv, b2, batch, pooled, N_NODES);

  // ---- head
  head_kernel<<<1, 128, 0, stream>>>(pooled, cnt, Wl, bl, (float*)d_out);
}